// RadialBasisArbitraryLayer_77386720740133
// MI455X (gfx1250) — compile-verified
//
#include <hip/hip_runtime.h>
#include <math.h>

#define HH 256
#define WW 256
#define NPT 160
#define NB 8
#define NITER (NPT / 4)
#define C_FACTOR 2.0f
#define BIG 1e10f

typedef __attribute__((ext_vector_type(2))) float v2f;
typedef __attribute__((ext_vector_type(8))) float v8f;

// ---------------- Kernel 1: support radius per batch ----------------
// c[b] = sqrt( max_i min_{j!=i} ||p_i - p_j||^2 ) * C_FACTOR
__global__ __launch_bounds__(256) void support_radius_kernel(
    const float* __restrict__ cpoint, float* __restrict__ c_out) {
  const int b = blockIdx.x;
  __shared__ float2 pts[NPT];
  __shared__ float red[256];
  const int t = threadIdx.x;
  if (t < NPT) {
    const float2* p2 = (const float2*)cpoint;
    pts[t] = p2[b * NPT + t];
  }
  __syncthreads();
  float mymin = 0.0f;  // inactive threads contribute 0 to the max
  if (t < NPT) {
    mymin = BIG;
    float2 p = pts[t];
    for (int j = 0; j < NPT; ++j) {
      float dx = p.x - pts[j].x;
      float dy = p.y - pts[j].y;
      float s = fmaf(dx, dx, dy * dy);
      if (j != t) mymin = fminf(mymin, s);
    }
  }
  red[t] = mymin;
  __syncthreads();
  for (int s = 128; s > 0; s >>= 1) {
    if (t < s) red[t] = fmaxf(red[t], red[t + s]);
    __syncthreads();
  }
  if (t == 0) c_out[b] = sqrtf(red[0]) * C_FACTOR;
}

// ---------------- Kernel 2: RBF field via VALU weights + WMMA accumulate ----
// Wave = 16 pixels (M). Loop points in chunks of 4 (K).
// A[16x4 f32]: lane L holds weights for pixel (L&15), points kb+2*(L>>4)+{0,1}
// B[4x16 f32]: col (L&15) = channel (only 0,1 nonzero), same K mapping
// D[16x16 f32]: row v+8*(L>>4) = pixel, col (L&15) = channel
//
// Coordinates are pre-scaled by 1/c, so the radicand is dx'^2+dy'^2 directly:
// one v_sqrt_f32 per weight, zero post/pre-scales in the hot loop.
__device__ __forceinline__ float wendland(float dx, float dy) {
  float d = __builtin_amdgcn_sqrtf(fmaf(dx, dx, dy * dy));
  float t1 = 1.0f - d;
  float t2 = t1 * t1;
  float w = (t2 * t2) * fmaf(4.0f, d, 1.0f);
  return (d < 1.0f) ? w : 0.0f;
}

__global__ __launch_bounds__(256) void rbf_field_kernel(
    const float* __restrict__ cpoint, const float* __restrict__ alpha,
    const float* __restrict__ cvals, float* __restrict__ out) {
  const int b = blockIdx.z;
  const int h = blockIdx.y;
  const int w_block = blockIdx.x * 128;

  __shared__ float4 s_pts4[NPT / 2];  // point pairs (2k,2k+1), pre-scaled by 1/c
  __shared__ float2 s_alp[NPT];       // raw alpha
  __shared__ v2f s_bv[NITER * 32];    // prebuilt per-lane B operands

  const int t = threadIdx.x;
  const float inv_c = 1.0f / cvals[b];
  if (t < NPT / 2) {
    float4 q = ((const float4*)cpoint)[b * (NPT / 2) + t];
    q.x *= inv_c; q.y *= inv_c; q.z *= inv_c; q.w *= inv_c;
    s_pts4[t] = q;
  }
  if (t < NPT) s_alp[t] = ((const float2*)alpha)[b * NPT + t];
  __syncthreads();

  // Build the B-operand stream: iteration it, lane ln needs
  //   bv = { alpha[k0][ch], alpha[k0+1][ch] } for ch = ln&15 < 2, else 0,
  //   k0 = 4*it + 2*(ln>>4).  Unconditional clamped loads + select: no
  //   exec-mask divergence anywhere near the hot loop.
  const float* s_alp_f = (const float*)s_alp;
  for (int i = t; i < NITER * 32; i += 256) {
    const int ln = i & 31;
    const int mm = ln & 15;
    const int k0 = ((i >> 5) << 2) + ((ln >> 4) << 1);
    const float v0 = s_alp_f[2 * k0 + (mm & 1)];
    const float v1 = s_alp_f[2 * (k0 + 1) + (mm & 1)];
    v2f bv;
    bv.x = (mm < 2) ? v0 : 0.0f;
    bv.y = (mm < 2) ? v1 : 0.0f;
    s_bv[i] = bv;
  }
  __syncthreads();

  const int wave = t >> 5;
  const int lane = t & 31;
  const int m = lane & 15;             // pixel (A row) / channel (B col)
  const int khalf = (lane >> 4) << 1;  // 0 or 2: K pair owned by this lane
  const int w0 = w_block + wave * 16;

  const float px = (float)(w0 + m) * inv_c;  // pre-scaled pixel coords
  const float py = (float)h * inv_c;

  v8f acc = {};
#pragma unroll 4
  for (int it = 0; it < NITER; ++it) {
    const float4 q = s_pts4[it * 2 + (khalf >> 1)];  // points k0, k0+1 (scaled)
    const v2f bv = s_bv[it * 32 + lane];

    v2f a;
    a.x = wendland(px - q.x, py - q.y);
    a.y = wendland(px - q.z, py - q.w);

    // D = A*B + C, all f32 — exact-precision accumulate on the matrix pipe,
    // co-executing with the VALU weight evaluation above.
    acc = __builtin_amdgcn_wmma_f32_16x16x4_f32(
        /*neg_a=*/false, a, /*neg_b=*/false, bv,
        /*c_mod=*/(short)0, acc, /*reuse_a=*/false, /*reuse_b=*/false);
  }

  // Only columns 0,1 of D are meaningful (channels). Lane L, VGPR v holds
  // pixel row = v + 8*(L>>4), channel = L&15.
  if (m < 2) {
    const int base = ((b * 2 + m) * HH + h) * WW + w0 + ((lane >> 4) << 3);
#pragma unroll
    for (int v = 0; v < 8; ++v) out[base + v] = acc[v];
  }
}

// ---------------- Launch ----------------
extern "C" void kernel_launch(void* const* d_in, const int* in_sizes, int n_in,
                              void* d_out, int out_size, void* d_ws,
                              size_t ws_size, hipStream_t stream) {
  const float* cpoint = (const float*)d_in[0];  // [8,160,2]
  const float* alpha  = (const float*)d_in[1];  // [8,160,2]
  float* out = (float*)d_out;                   // [8,2,256,256]
  float* cvals = (float*)d_ws;                  // [8] support radii

  support_radius_kernel<<<NB, 256, 0, stream>>>(cpoint, cvals);

  dim3 grid(WW / 128, HH, NB);
  rbf_field_kernel<<<grid, 256, 0, stream>>>(cpoint, alpha, cvals, out);
}